// RejectionSampler_70394513981673
// MI455X (gfx1250) — compile-verified
//
#include <hip/hip_runtime.h>
#include <math.h>

// ---------------------------------------------------------------------------
// Greedy speculative-decode rejection sampler for MI455X (gfx1250, wave32).
//
// Stage 1: per-token-row argmax over vocab. Pure streaming reduction:
//   512 MB single pass -> non-temporal global_load_b128 (5 in flight/lane),
//   global_prefetch_b8 ahead of the stream, branchless cmp/cndmask updates,
//   wave32 shfl_xor butterfly + LDS cross-wave combine.
// Stage 2: per-batch-row accept scan (tiny, branchless).
// ---------------------------------------------------------------------------

typedef float v4f __attribute__((ext_vector_type(4)));

#define WAVE 32
#define BLK  256   // 8 wave32 waves per workgroup
#define CHUNK 5    // float4 loads in flight per lane

__device__ __forceinline__ void upd4(v4f v, int base, float& best, int& bestIdx) {
  #pragma unroll
  for (int c = 0; c < 4; ++c) {
    const float val = v[c];
    const int   idx = base + c;
    // Branchless: lowers to v_cmp_gt_f32 + v_cndmask pair. Strict '>' keeps
    // the earliest index (scan order is increasing per lane).
    const bool gt = val > best;
    best    = gt ? val : best;
    bestIdx = gt ? idx : bestIdx;
  }
}

__global__ __launch_bounds__(BLK) void argmax_rows_kernel(
    const float* __restrict__ logits, int vocab, int* __restrict__ row_argmax) {
  const int row = blockIdx.x;
  const float* __restrict__ p = logits + (long long)row * (long long)vocab;
  const int t = threadIdx.x;

  float best = -__builtin_inff();
  int bestIdx = vocab;  // sentinel larger than any valid index

  const int stride = BLK * 4;                  // 1024 floats per block iter
  const int nIter  = vocab / stride;           // 125 for vocab=128000
  const int vecEnd = nIter * stride;

  int base = t * 4;
  int it = 0;

  // Main loop: CHUNK back-to-back NT b128 loads (memory-level parallelism),
  // one speculative prefetch ~40KB ahead, then straight-line VALU updates.
  for (; it + CHUNK <= nIter; it += CHUNK) {
    v4f v[CHUNK];
    #pragma unroll
    for (int u = 0; u < CHUNK; ++u) {
      v[u] = __builtin_nontemporal_load(
          reinterpret_cast<const v4f*>(p + base + u * stride));
    }
    __builtin_prefetch(p + base + 2 * CHUNK * stride, 0, 1);  // global_prefetch_b8
    #pragma unroll
    for (int u = 0; u < CHUNK; ++u) {
      upd4(v[u], base + u * stride, best, bestIdx);
    }
    base += CHUNK * stride;
  }
  // leftover full iterations (none for vocab=128000)
  for (; it < nIter; ++it) {
    v4f v = __builtin_nontemporal_load(reinterpret_cast<const v4f*>(p + base));
    upd4(v, base, best, bestIdx);
    base += stride;
  }
  // scalar tail (empty when vocab % 1024 == 0); tail indices exceed all seen.
  for (int i = vecEnd + t; i < vocab; i += BLK) {
    const float val = p[i];
    const bool gt = val > best;
    best    = gt ? val : best;
    bestIdx = gt ? i   : bestIdx;
  }

  // wave32 butterfly reduction, tie-break toward lower index (branchless)
  #pragma unroll
  for (int off = WAVE / 2; off >= 1; off >>= 1) {
    const float ov = __shfl_xor(best, off, WAVE);
    const int   oi = __shfl_xor(bestIdx, off, WAVE);
    const bool take = (ov > best) || (ov == best && oi < bestIdx);
    best    = take ? ov : best;
    bestIdx = take ? oi : bestIdx;
  }

  // cross-wave combine (8 waves) through LDS
  __shared__ float s_val[BLK / WAVE];
  __shared__ int   s_idx[BLK / WAVE];
  const int wave = t / WAVE;
  if ((t & (WAVE - 1)) == 0) { s_val[wave] = best; s_idx[wave] = bestIdx; }
  __syncthreads();
  if (t == 0) {
    #pragma unroll
    for (int w = 1; w < BLK / WAVE; ++w) {
      const float ov = s_val[w];
      const int   oi = s_idx[w];
      const bool take = (ov > best) || (ov == best && oi < bestIdx);
      best    = take ? ov : best;
      bestIdx = take ? oi : bestIdx;
    }
    row_argmax[row] = bestIdx;
  }
}

// Stage 2: accept = cumprod(out[:, :k] == spec); emit argmax tokens for the
// accepted prefix plus the first rejected/bonus slot, INVALID (-1) elsewhere.
__global__ void accept_scan_kernel(const int* __restrict__ row_argmax,
                                   const int* __restrict__ spec_ids,
                                   float* __restrict__ out, int B) {
  const int b = blockIdx.x * blockDim.x + threadIdx.x;
  if (b >= B) return;
  const int k = 7;                 // spec_len (reference fixes k=7, k+1=8)

  int o[8];
  #pragma unroll
  for (int i = 0; i < 8; ++i) o[i] = row_argmax[b * 8 + i];

  // r = number of leading draft tokens matching the target argmax
  int r = 0;
  bool acc = true;
  #pragma unroll
  for (int i = 0; i < k; ++i) {
    acc = acc && (o[i] == spec_ids[b * k + i]);
    r += acc ? 1 : 0;
  }

  // emit positions 0..r inclusive (accepted prefix + first-zero/bonus slot)
  #pragma unroll
  for (int i = 0; i < 8; ++i) {
    out[b * 8 + i] = (i <= r) ? (float)o[i] : -1.0f;
  }
}

extern "C" void kernel_launch(void* const* d_in, const int* in_sizes, int n_in,
                              void* d_out, int out_size, void* d_ws, size_t ws_size,
                              hipStream_t stream) {
  const float* logits = (const float*)d_in[0];
  const int*   spec   = (const int*)d_in[1];
  float*       out    = (float*)d_out;

  const int nSpec = in_sizes[1];       // B * 7
  const int B     = nSpec / 7;         // 128
  const int rows  = B * 8;             // 1024 token rows
  const int vocab = (int)((long long)in_sizes[0] / rows);  // 128000

  int* row_argmax = (int*)d_ws;        // rows * sizeof(int) scratch

  argmax_rows_kernel<<<rows, BLK, 0, stream>>>(logits, vocab, row_argmax);
  accept_scan_kernel<<<(B + 127) / 128, 128, 0, stream>>>(row_argmax, spec, out, B);
}